// VideoGNN_48369921687740
// MI455X (gfx1250) — compile-verified
//
#include <hip/hip_runtime.h>
#include <hip/hip_bf16.h>

typedef float v2f __attribute__((ext_vector_type(2)));
typedef float v8f __attribute__((ext_vector_type(8)));

#define NROW 12544
#define NPTS 196
#define CDIM 256

#define FLAG_RELU   1
#define FLAG_ADDTMP 2
#define FLAG_ROLLA  4

__device__ __forceinline__ v8f wmma4(v2f a, v2f b, v8f c) {
  // V_WMMA_F32_16X16X4_F32: D(16x16,f32) = A(16x4,f32) * B(4x16,f32) + C
  return __builtin_amdgcn_wmma_f32_16x16x4_f32(false, a, false, b, (short)0, c, false, false);
}

__device__ __forceinline__ int roll_batch(int b) {
  // T = 16: y[t] = x[t+1 mod T] within each group of 16 batches
  return (b & ~15) | ((b + 1) & 15);
}

// ---------------------------------------------------------------------------
// Transpose x (b,c,n) -> x_sp (b*n, c) and squared norms per point
// ---------------------------------------------------------------------------
__global__ __launch_bounds__(256) void k_transpose_sqn(
    const float* __restrict__ x, float* __restrict__ xsp, float* __restrict__ sqn) {
  int row = blockIdx.x;               // b*196 + i
  int b = row / NPTS, i = row - b * NPTS;
  int c = threadIdx.x;                // 0..255
  float v = x[((size_t)b * CDIM + c) * NPTS + i];
  xsp[(size_t)row * CDIM + c] = v;
  __shared__ float red[256];
  red[c] = v * v;
  __syncthreads();
  for (int s = 128; s > 0; s >>= 1) {
    if (c < s) red[c] += red[c + s];
    __syncthreads();
  }
  if (c == 0) sqn[row] = red[0];
}

// ---------------------------------------------------------------------------
// KNN: one wave per (batch, 16-row tile). WMMA gram (2 interleaved K-chains)
// + top-9 insertion select per row.
// ---------------------------------------------------------------------------
__global__ __launch_bounds__(32) void k_knn(
    const float* __restrict__ xsp, const float* __restrict__ sqn,
    int* __restrict__ nbr, int roll) {
  int b = blockIdx.y;
  int rt = blockIdx.x;                 // 13 row tiles (196 -> 208)
  int kb = roll ? roll_batch(b) : b;
  int lane = threadIdx.x;
  int half = lane >> 4, l16 = lane & 15;

  __shared__ float dist[16 * 208];

  const float* Q  = xsp + (size_t)b  * NPTS * CDIM;
  const float* Kp = xsp + (size_t)kb * NPTS * CDIM;
  const float* qs = sqn + (size_t)b  * NPTS;
  const float* ks = sqn + (size_t)kb * NPTS;

  int qrow = rt * 16 + l16;
  int qrc = qrow < NPTS ? qrow : NPTS - 1;
  float qsv[8];
#pragma unroll
  for (int r = 0; r < 8; r++) {
    int m = rt * 16 + r + 8 * half;
    qsv[r] = (m < NPTS) ? qs[m] : 0.0f;
  }
  const float* Ap = Q + (size_t)qrc * CDIM + 2 * half;

  for (int ct = 0; ct < 13; ++ct) {
    int kcol = ct * 16 + l16;
    int kcc = kcol < NPTS ? kcol : NPTS - 1;
    const float* Bp = Kp + (size_t)kcc * CDIM + 2 * half;
    v8f acc0 = {}, acc1 = {};
    for (int k = 0; k < CDIM; k += 8) {
      v2f a0 = *(const v2f*)(Ap + k);
      v2f b0 = *(const v2f*)(Bp + k);
      v2f a1 = *(const v2f*)(Ap + k + 4);
      v2f b1 = *(const v2f*)(Bp + k + 4);
      acc0 = wmma4(a0, b0, acc0);
      acc1 = wmma4(a1, b1, acc1);
    }
    float ksv = ks[kcc];
    bool cval = kcol < NPTS;
#pragma unroll
    for (int r = 0; r < 8; r++) {
      int m = r + 8 * half;
      float dot = acc0[r] + acc1[r];
      float d = cval ? (qsv[r] - 2.0f * dot + ksv) : 3.0e38f;
      dist[m * 208 + kcol] = d;
    }
  }
  __syncthreads();

  if (lane < 16) {
    int row = rt * 16 + lane;
    if (row < NPTS) {
      float best[9]; int bidx[9];
#pragma unroll
      for (int s = 0; s < 9; s++) { best[s] = 3.0e38f; bidx[s] = 0; }
      for (int j = 0; j < NPTS; j++) {
        float d = dist[lane * 208 + j];
        if (d < best[8]) {
          int p = 8;
          while (p > 0 && d < best[p - 1]) {
            best[p] = best[p - 1]; bidx[p] = bidx[p - 1]; --p;
          }
          best[p] = d; bidx[p] = j;
        }
      }
      int* out = nbr + ((size_t)(b * NPTS + row)) * 16;
#pragma unroll
      for (int s = 0; s < 9; s++) out[s] = bidx[s];
    }
  }
}

// ---------------------------------------------------------------------------
// Generic WMMA GEMM: D = epilogue(A[12544 x K] @ W[K x N]).
// Each wave computes a 16x64 tile: 4 independent accumulators reusing one
// A fragment per K-step (4x less A traffic, WMMA dep distance 4).
// 8 waves/block -> 128x64 block tile; grid = (ceil(N/64), 98).
// Epilogue: + bias; optionally + deg[m]*tmpIn[m][n]; optional ReLU.
// FLAG_ROLLA remaps A rows by the batch roll (y_feat view of x_feat).
// ---------------------------------------------------------------------------
__global__ __launch_bounds__(256) void k_gemm(
    const float* __restrict__ A, const float* __restrict__ W,
    const float* __restrict__ bias, const float* __restrict__ tmpIn,
    const float* __restrict__ degv, float* __restrict__ D,
    int K, int N, int lda, int ldd, int ldtmp, int flags) {
  int lane = threadIdx.x & 31;
  int wave = threadIdx.x >> 5;
  int col0 = blockIdx.x * 64;
  int row0 = (blockIdx.y * 8 + wave) * 16;
  int half = lane >> 4, l16 = lane & 15;

  int arow = row0 + l16;
  if (flags & FLAG_ROLLA) {
    int b = arow / NPTS, i = arow - b * NPTS;
    arow = roll_batch(b) * NPTS + i;
  }
  const float* Ap = A + (size_t)arow * lda + 2 * half;

  const float* Wp[4];
#pragma unroll
  for (int t = 0; t < 4; t++) {
    int n = col0 + 16 * t + l16;
    Wp[t] = W + ((n < N) ? n : 0);     // clamp; garbage cols never stored
  }

  v8f acc[4] = {{}, {}, {}, {}};
  for (int k = 0; k < K; k += 4) {
    if ((k & 31) == 0) __builtin_prefetch(Ap + k + 32, 0, 0);
    v2f a = *(const v2f*)(Ap + k);
    size_t kk = (size_t)(k + 2 * half);
#pragma unroll
    for (int t = 0; t < 4; t++) {
      v2f bv;
      bv.x = Wp[t][kk * N];
      bv.y = Wp[t][(kk + 1) * N];
      acc[t] = wmma4(a, bv, acc[t]);
    }
  }

#pragma unroll
  for (int t = 0; t < 4; t++) {
    int n = col0 + 16 * t + l16;
    if (n < N) {
      float bi = bias ? bias[n] : 0.0f;
#pragma unroll
      for (int r = 0; r < 8; r++) {
        int m = row0 + r + 8 * half;
        float v = acc[t][r] + bi;
        if (flags & FLAG_ADDTMP) {
          float dg = degv ? degv[m] : 1.0f;
          v += dg * tmpIn[(size_t)m * ldtmp + n];
        }
        if (flags & FLAG_RELU) v = fmaxf(v, 0.0f);
        D[(size_t)m * ldd + n] = v;
      }
    }
  }
}

// ---------------------------------------------------------------------------
// Head: mu(64->32) -> logits(32->9) -> argmax -> validf mask + deg.
// One wave per row, 8 rows per block.
// ---------------------------------------------------------------------------
__global__ __launch_bounds__(256) void k_head(
    const float* __restrict__ h64,
    const float* __restrict__ mu_w, const float* __restrict__ mu_b,
    const float* __restrict__ dec_w, const float* __restrict__ dec_b,
    float* __restrict__ vf, float* __restrict__ deg) {
  int lane = threadIdx.x & 31;
  int wv = threadIdx.x >> 5;
  int row = blockIdx.x * 8 + wv;
  __shared__ float smu[8][32];
  __shared__ float slog[8][16];

  const float* h = h64 + (size_t)row * 64;
  float m = mu_b[lane];
  for (int k = 0; k < 64; k++) m += h[k] * mu_w[k * 32 + lane];
  smu[wv][lane] = m;
  __syncthreads();

  if (lane < 9) {
    float v = dec_b[lane];
    for (int k = 0; k < 32; k++) v += smu[wv][k] * dec_w[k * 9 + lane];
    slog[wv][lane] = v;
  }
  __syncthreads();

  if (lane == 0) {
    float bv = slog[wv][0]; int bi = 0;
    for (int j = 1; j < 9; j++) {
      float v = slog[wv][j];
      if (v > bv) { bv = v; bi = j; }      // first max wins (jnp.argmax)
    }
    deg[row] = (float)bi;
    float* v = vf + (size_t)row * 16;
#pragma unroll
    for (int k = 0; k < 16; k++) v[k] = (k < bi) ? 1.0f : 0.0f;
  }
}

// ---------------------------------------------------------------------------
// Edge message: em[r][c] = sum_k vf[r][k]*feat[src(nbr)][c] - deg[r]*feat[src_r][c]
// ---------------------------------------------------------------------------
__global__ __launch_bounds__(256) void k_edge_gather(
    const float* __restrict__ feat, const int* __restrict__ nbr,
    const float* __restrict__ vf, const float* __restrict__ degv,
    float* __restrict__ em, int roll) {
  int row = blockIdx.x;
  int c = threadIdx.x;
  int b = row / NPTS, i = row - b * NPTS;
  int base = (roll ? roll_batch(b) : b) * NPTS;
  float dg = degv[row];
  float acc = -dg * feat[(size_t)(base + i) * CDIM + c];
  const int* nb = nbr + (size_t)row * 16;
  const float* v = vf + (size_t)row * 16;
#pragma unroll
  for (int k = 0; k < 9; k++)
    acc += v[k] * feat[(size_t)(base + nb[k]) * CDIM + c];
  em[(size_t)row * CDIM + c] = acc;
}

// ---------------------------------------------------------------------------
// scores[r] = [x_agg | y_agg] . attn_w + attn_b
// ---------------------------------------------------------------------------
__global__ __launch_bounds__(256) void k_attn(
    const float* __restrict__ xa, const float* __restrict__ ya,
    const float* __restrict__ aw, const float* __restrict__ ab,
    float* __restrict__ scores) {
  int row = blockIdx.x;
  int t = threadIdx.x;
  const float* xr = xa + (size_t)row * 512;
  const float* yr = ya + (size_t)row * 512;
  float s = xr[t] * aw[t] + xr[t + 256] * aw[t + 256] +
            yr[t] * aw[512 + t] + yr[t + 256] * aw[768 + t];
  __shared__ float red[256];
  red[t] = s;
  __syncthreads();
  for (int st = 128; st > 0; st >>= 1) {
    if (t < st) red[t] += red[t + st];
    __syncthreads();
  }
  if (t == 0) scores[row] = red[0] + ab[0];
}

// Global softmax over 12544 rows: compute [max, sum] with one block.
__global__ __launch_bounds__(256) void k_softmax_reduce(
    const float* __restrict__ scores, float* __restrict__ red2) {
  int t = threadIdx.x;
  __shared__ float red[256];
  float mx = -3.0e38f;
  for (int i = t; i < NROW; i += 256) mx = fmaxf(mx, scores[i]);
  red[t] = mx;
  __syncthreads();
  for (int s = 128; s > 0; s >>= 1) {
    if (t < s) red[t] = fmaxf(red[t], red[t + s]);
    __syncthreads();
  }
  float gmax = red[0];
  __syncthreads();
  float sm = 0.0f;
  for (int i = t; i < NROW; i += 256) sm += __expf(scores[i] - gmax);
  red[t] = sm;
  __syncthreads();
  for (int s = 128; s > 0; s >>= 1) {
    if (t < s) red[t] += red[t + s];
    __syncthreads();
  }
  if (t == 0) { red2[0] = gmax; red2[1] = red[0]; }
}

// fused (into x_agg in place): w*x_agg + (1-w)*y_agg
__global__ __launch_bounds__(256) void k_fuse(
    float* __restrict__ xa, const float* __restrict__ ya,
    const float* __restrict__ scores, const float* __restrict__ red2) {
  int row = blockIdx.x;
  int t = threadIdx.x;
  float w = __expf(scores[row] - red2[0]) / red2[1];
  float* xr = xa + (size_t)row * 512;
  const float* yr = ya + (size_t)row * 512;
  xr[t]       = w * xr[t]       + (1.0f - w) * yr[t];
  xr[t + 256] = w * xr[t + 256] + (1.0f - w) * yr[t + 256];
}

// out (b, 512, 196) <- upd (b*196, 512)
__global__ __launch_bounds__(224) void k_writeout(
    const float* __restrict__ upd, float* __restrict__ out) {
  int o = blockIdx.x;      // 512
  int b = blockIdx.y;      // 64
  int i = threadIdx.x;
  if (i < NPTS)
    out[((size_t)b * 512 + o) * NPTS + i] = upd[(size_t)(b * NPTS + i) * 512 + o];
}

// ---------------------------------------------------------------------------
static inline void gemm(hipStream_t s, const float* A, int lda,
                        const float* W, const float* bias,
                        const float* tmpIn, int ldtmp, const float* degv,
                        float* D, int ldd, int K, int N, int flags) {
  dim3 grid((N + 63) / 64, NROW / 128);
  k_gemm<<<grid, 256, 0, s>>>(A, W, bias, tmpIn, degv, D, K, N, lda, ldd, ldtmp, flags);
}

extern "C" void kernel_launch(void* const* d_in, const int* in_sizes, int n_in,
                              void* d_out, int out_size, void* d_ws, size_t ws_size,
                              hipStream_t stream) {
  (void)in_sizes; (void)n_in; (void)out_size; (void)ws_size;
  const float* x       = (const float*)d_in[0];   // (64,256,14,14) == x_feat (12544,256)
  const float* fc1_w   = (const float*)d_in[1];
  const float* fc1_b   = (const float*)d_in[2];
  const float* fc2_w   = (const float*)d_in[3];
  const float* fc2_b   = (const float*)d_in[4];
  const float* fce_w   = (const float*)d_in[5];
  const float* fce_b   = (const float*)d_in[6];
  const float* inout_w = (const float*)d_in[7];
  const float* inout_b = (const float*)d_in[8];
  const float* attn_w  = (const float*)d_in[9];
  const float* attn_b  = (const float*)d_in[10];
  const float* upd_w   = (const float*)d_in[11];
  const float* upd_b   = (const float*)d_in[12];
  const float* kp_w    = (const float*)d_in[13];
  const float* kp_b    = (const float*)d_in[14];
  const float* enc_w   = (const float*)d_in[15];
  const float* enc_b   = (const float*)d_in[16];
  const float* mu_w    = (const float*)d_in[17];
  const float* mu_b    = (const float*)d_in[18];
  const float* dec_w   = (const float*)d_in[19];
  const float* dec_b   = (const float*)d_in[20];
  float* out = (float*)d_out;

  // ---- workspace layout (floats) ----
  float* ws = (float*)d_ws;
  size_t o = 0;
  float* xsp    = ws + o; o += (size_t)NROW * CDIM;   // x_sp; later em / em2
  float* sqn    = ws + o; o += NROW;
  float* deg    = ws + o; o += NROW;
  float* scores = ws + o; o += NROW;
  float* red2   = ws + o; o += 16;
  float* vf     = ws + o; o += (size_t)NROW * 16;
  float* h64    = ws + o; o += (size_t)NROW * 64;
  float* bufA   = ws + o; o += (size_t)NROW * 512;    // h500 -> x_agg -> fused
  float* bufB   = ws + o; o += (size_t)NROW * CDIM;   // tmp1 / tmp2
  float* bufC   = ws + o; o += (size_t)NROW * 512;    // h1|h2 -> upd
  float* bufD   = ws + o; o += (size_t)NROW * 512;    // y_agg -> xin
  int* xnbr = (int*)(ws + o); o += (size_t)NROW * 16;
  int* ynbr = (int*)(ws + o); o += (size_t)NROW * 16;

  float* em = xsp;                 // alias: x_sp dead after knn
  float* h1 = bufC;
  float* h2 = bufC + (size_t)NROW * CDIM;

  // 1) transpose + squared norms
  k_transpose_sqn<<<NROW, 256, 0, stream>>>(x, xsp, sqn);

  // 2) KNN (x->x and x->rolled x)
  k_knn<<<dim3(13, 64), 32, 0, stream>>>(xsp, sqn, xnbr, 0);
  k_knn<<<dim3(13, 64), 32, 0, stream>>>(xsp, sqn, ynbr, 1);

  // 3) keypoint MLP: h500 = x@kp_w + b ; h64 = relu(h500@enc_w + b)
  gemm(stream, x, CDIM, kp_w, kp_b, nullptr, 0, nullptr, bufA, 512, 256, 500, 0);
  gemm(stream, bufA, 512, enc_w, enc_b, nullptr, 0, nullptr, h64, 64, 500, 64, FLAG_RELU);

  // 4) mu -> logits -> argmax -> validf + deg
  k_head<<<NROW / 8, 256, 0, stream>>>(h64, mu_w, mu_b, dec_w, dec_b, vf, deg);

  const float* wb1 = fc1_w + 256 * 256;
  const float* wb2 = fc2_w + 256 * 256;

  // 5) edge_agg on (x_feat, x_nbr) -> x_agg in bufA
  k_edge_gather<<<NROW, 256, 0, stream>>>(x, xnbr, vf, deg, em, 0);
  gemm(stream, x, CDIM, fc1_w, fc1_b, nullptr, 0, nullptr, bufB, CDIM, 256, 256, 0);
  gemm(stream, em, CDIM, wb1, nullptr, bufB, CDIM, deg, h1, CDIM, 256, 256, FLAG_RELU | FLAG_ADDTMP);
  k_edge_gather<<<NROW, 256, 0, stream>>>(h1, xnbr, vf, deg, em, 0);
  gemm(stream, h1, CDIM, fc2_w, fc2_b, nullptr, 0, nullptr, bufB, CDIM, 256, 256, 0);
  gemm(stream, em, CDIM, wb2, nullptr, bufB, CDIM, deg, h2, CDIM, 256, 256, FLAG_ADDTMP);
  gemm(stream, h2, CDIM, fce_w, fce_b, nullptr, 0, nullptr, bufA, 512, 256, 512, 0);

  // 6) edge_agg on (y_feat = rolled x_feat, y_nbr) -> y_agg in bufD
  k_edge_gather<<<NROW, 256, 0, stream>>>(x, ynbr, vf, deg, em, 1);
  gemm(stream, x, CDIM, fc1_w, fc1_b, nullptr, 0, nullptr, bufB, CDIM, 256, 256, FLAG_ROLLA);
  gemm(stream, em, CDIM, wb1, nullptr, bufB, CDIM, deg, h1, CDIM, 256, 256, FLAG_RELU | FLAG_ADDTMP);
  k_edge_gather<<<NROW, 256, 0, stream>>>(h1, ynbr, vf, deg, em, 0);
  gemm(stream, h1, CDIM, fc2_w, fc2_b, nullptr, 0, nullptr, bufB, CDIM, 256, 256, 0);
  gemm(stream, em, CDIM, wb2, nullptr, bufB, CDIM, deg, h2, CDIM, 256, 256, FLAG_ADDTMP);
  gemm(stream, h2, CDIM, fce_w, fce_b, nullptr, 0, nullptr, bufD, 512, 256, 512, 0);

  // 7) attention + global softmax + fuse (fused -> bufA in place)
  k_attn<<<NROW, 256, 0, stream>>>(bufA, bufD, attn_w, attn_b, scores);
  k_softmax_reduce<<<1, 256, 0, stream>>>(scores, red2);
  k_fuse<<<NROW, 256, 0, stream>>>(bufA, bufD, scores, red2);

  // 8) xin = x@inout_w + b (y_agg dead -> bufD)
  gemm(stream, x, CDIM, inout_w, inout_b, nullptr, 0, nullptr, bufD, 512, 256, 512, 0);

  // 9) upd = relu(xin@upd_w[:512] + fused@upd_w[512:] + b)  -> bufC
  gemm(stream, bufD, 512, upd_w, upd_b, nullptr, 0, nullptr, bufC, 512, 512, 512, 0);
  gemm(stream, bufA, 512, upd_w + 512 * 512, nullptr, bufC, 512, nullptr, bufC, 512, 512, 512,
       FLAG_RELU | FLAG_ADDTMP);

  // 10) transpose to (b, 512, 196)
  k_writeout<<<dim3(512, 64), 224, 0, stream>>>(bufC, out);
}